// GATE_55121610277321
// MI455X (gfx1250) — compile-verified
//
#include <hip/hip_runtime.h>
#include <math.h>

typedef __attribute__((ext_vector_type(2))) float v2f;
typedef __attribute__((ext_vector_type(8))) float v8f;

#define N_NODES 20000
#define N_EDGES 160000
#define IN_DIM  3000
#define F1      512
#define F2      30
#define E_TOT   (N_EDGES + N_NODES)
#define NEG_SLOPE 0.2f
#define EPSV    1e-12f

// ---------------------------------------------------------------------------
// fp32 WMMA GEMM: C[M,N] = A[M,K] @ B   (TRANS_B ? B:[N,K] : B:[K,N])
// Block tile 64(M) x 128(N) x 32(K); 8 waves; each wave: 16x64 via 4 wmma frags.
// Double-buffered LDS; As[m][k] and Bs[n][k] layouts so both A and B fragments
// are single contiguous ds_load_b64 into even-aligned VGPR pairs.
// ---------------------------------------------------------------------------
template <bool TRANS_B>
__global__ __launch_bounds__(256) void gat_gemm_wmma(
    const float* __restrict__ A, const float* __restrict__ B,
    float* __restrict__ C, int M, int N, int K)
{
    constexpr int BM = 64, BN = 128, BK = 32, LDK = BK + 2;
    __shared__ float As[2][BM][LDK];    // [buf][m][k]
    __shared__ float Bs[2][BN][LDK];    // [buf][n][k]  (k contiguous!)

    const int m0   = blockIdx.y * BM;
    const int n0   = blockIdx.x * BN;
    const int t    = threadIdx.x;
    const int lane = t & 31;
    const int wave = t >> 5;
    const int wm   = (wave & 3) * 16;       // wave's M offset in tile
    const int wn   = (wave >> 2) * 64;      // wave's N offset in tile

    const int mfrag = lane & 15;            // lane index inside 16
    const int khalf = (lane >> 4) * 2;      // lanes 0-15 -> k0,k1 ; 16-31 -> k2,k3

    v8f acc[4] = {};
    float ra[8];
    float rb[16];

    const int nchunks = (K + BK - 1) / BK;

    auto loadA = [&](int k0) {
        #pragma unroll
        for (int i = 0; i < 8; ++i) {
            int idx = t + i * 256;
            int r = idx >> 5, c = idx & 31;
            int gm = m0 + r, gk = k0 + c;
            ra[i] = (gm < M && gk < K) ? A[(size_t)gm * K + gk] : 0.0f;
        }
    };
    auto commitA = [&](int b) {
        #pragma unroll
        for (int i = 0; i < 8; ++i) {
            int idx = t + i * 256;
            As[b][idx >> 5][idx & 31] = ra[i];
        }
    };
    auto loadB = [&](int k0) {
        #pragma unroll
        for (int i = 0; i < 16; ++i) {
            int idx = t + i * 256;
            if (!TRANS_B) {
                int k = idx >> 7, n = idx & 127;          // coalesced over n
                int gk = k0 + k, gn = n0 + n;
                rb[i] = (gk < K && gn < N) ? B[(size_t)gk * N + gn] : 0.0f;
            } else {
                int k = idx & 31, n = idx >> 5;           // coalesced over k
                int gk = k0 + k, gn = n0 + n;
                rb[i] = (gn < N && gk < K) ? B[(size_t)gn * K + gk] : 0.0f;
            }
        }
    };
    auto commitB = [&](int b) {
        #pragma unroll
        for (int i = 0; i < 16; ++i) {
            int idx = t + i * 256;
            if (!TRANS_B) Bs[b][idx & 127][idx >> 7] = rb[i];
            else          Bs[b][idx >> 5][idx & 31]  = rb[i];
        }
    };

    // prologue: stage chunk 0
    loadA(0); loadB(0);
    commitA(0); commitB(0);
    __syncthreads();

    for (int c = 0; c < nchunks; ++c) {
        const int cur = c & 1;
        if (c + 1 < nchunks) {                 // issue next-chunk global loads;
            loadA((c + 1) * BK);               // VMEM overlaps the WMMAs below
            loadB((c + 1) * BK);
        }
        const float* ap = &As[cur][wm + mfrag][khalf];
        #pragma unroll
        for (int kk = 0; kk < BK; kk += 4) {
            v2f a;
            a.x = ap[kk];
            a.y = ap[kk + 1];
            #pragma unroll
            for (int f = 0; f < 4; ++f) {
                const float* bp = &Bs[cur][wn + f * 16 + mfrag][kk + khalf];
                v2f b;
                b.x = bp[0];
                b.y = bp[1];
                acc[f] = __builtin_amdgcn_wmma_f32_16x16x4_f32(
                    false, a, false, b, (short)0, acc[f], false, false);
            }
        }
        __syncthreads();
        if (c + 1 < nchunks) {
            const int nb = (c + 1) & 1;
            commitA(nb); commitB(nb);
            __syncthreads();
        }
    }

    // store: C/D layout: vgpr r, lanes 0-15 -> M=r ; lanes 16-31 -> M=r+8
    const int mBase = m0 + wm + ((lane >> 4) << 3);
    #pragma unroll
    for (int f = 0; f < 4; ++f) {
        int gn = n0 + wn + f * 16 + (lane & 15);
        if (gn < N) {
            #pragma unroll
            for (int r = 0; r < 8; ++r) {
                int gm = mBase + r;
                if (gm < M) C[(size_t)gm * N + gn] = acc[f][r];
            }
        }
    }
}

// ---------------------------------------------------------------------------
// per-node attention logits: a_src[i] = x1[i,:].att_src ; a_dst likewise
// one wave per node
// ---------------------------------------------------------------------------
__global__ void gat_node_dots(const float* __restrict__ x1,
                              const float* __restrict__ attS,
                              const float* __restrict__ attD,
                              float* __restrict__ aS, float* __restrict__ aD)
{
    int gid  = blockIdx.x * blockDim.x + threadIdx.x;
    int wid  = gid >> 5;
    int lane = gid & 31;
    if (wid >= N_NODES) return;
    const float* row = x1 + (size_t)wid * F1;
    float s1 = 0.f, s2 = 0.f;
    for (int f = lane; f < F1; f += 32) {
        float v = row[f];
        s1 += v * attS[f];
        s2 += v * attD[f];
    }
    for (int o = 16; o > 0; o >>= 1) {
        s1 += __shfl_xor(s1, o, 32);
        s2 += __shfl_xor(s2, o, 32);
    }
    if (lane == 0) { aS[wid] = s1; aD[wid] = s2; }
}

__global__ void gat_fill_f32(float* p, float v, int n)
{
    int i = blockIdx.x * blockDim.x + threadIdx.x;
    if (i < n) p[i] = v;
}

__device__ __forceinline__ void gat_edge_sd(const int* __restrict__ ei, int e,
                                            int& s, int& d)
{
    if (e < N_EDGES) { s = ei[e]; d = ei[N_EDGES + e]; }
    else             { s = d = e - N_EDGES; }      // self loops
}

__device__ __forceinline__ void gat_atomicMaxF(float* addr, float val)
{
    if (val >= 0.0f) atomicMax((int*)addr, __float_as_int(val));
    else             atomicMin((unsigned int*)addr, (unsigned int)__float_as_int(val));
}

// pass 1: leaky-relu logit + segment max over dst
__global__ void gat_edge_logit_max(const int* __restrict__ ei,
                                   const float* __restrict__ aS,
                                   const float* __restrict__ aD,
                                   float* __restrict__ logit,
                                   float* __restrict__ segmax)
{
    int e = blockIdx.x * blockDim.x + threadIdx.x;
    if (e >= E_TOT) return;
    int s, d; gat_edge_sd(ei, e, s, d);
    float x = aS[s] + aD[d];
    x = x > 0.0f ? x : NEG_SLOPE * x;
    logit[e] = x;
    gat_atomicMaxF(&segmax[d], x);
}

// pass 2: exp(logit - max[dst]) + segment sum
__global__ void gat_edge_exp_sum(const int* __restrict__ ei,
                                 float* __restrict__ logit,    // in: logit, out: e
                                 const float* __restrict__ segmax,
                                 float* __restrict__ segsum)
{
    int e = blockIdx.x * blockDim.x + threadIdx.x;
    if (e >= E_TOT) return;
    int s, d; gat_edge_sd(ei, e, s, d);
    float ev = expf(logit[e] - segmax[d]);
    logit[e] = ev;
    unsafeAtomicAdd(&segsum[d], ev);
}

// pass 3: alpha = e / (sum[dst] + eps)
__global__ void gat_edge_alpha(const int* __restrict__ ei,
                               float* __restrict__ alpha,
                               const float* __restrict__ segsum)
{
    int e = blockIdx.x * blockDim.x + threadIdx.x;
    if (e >= E_TOT) return;
    int s, d; gat_edge_sd(ei, e, s, d);
    alpha[e] = alpha[e] / (segsum[d] + EPSV);
}

// scatter aggregate: out[dst,:] += x[src,:] * alpha[e]; one wave per edge
__global__ void gat_propagate(const int* __restrict__ ei,
                              const float* __restrict__ x,
                              const float* __restrict__ alpha,
                              float* __restrict__ out)
{
    int gid  = blockIdx.x * blockDim.x + threadIdx.x;
    int wid  = gid >> 5;
    int lane = gid & 31;
    if (wid >= E_TOT) return;
    int s, d; gat_edge_sd(ei, wid, s, d);
    float a = alpha[wid];
    const float4* xs = (const float4*)(x + (size_t)s * F1);
    float* od = out + (size_t)d * F1;
    #pragma unroll
    for (int i = 0; i < F1 / 128; ++i) {           // 4 iters of float4 x 32 lanes
        float4 v = xs[lane + i * 32];
        int base = (lane + i * 32) * 4;
        unsafeAtomicAdd(&od[base + 0], v.x * a);
        unsafeAtomicAdd(&od[base + 1], v.y * a);
        unsafeAtomicAdd(&od[base + 2], v.z * a);
        unsafeAtomicAdd(&od[base + 3], v.w * a);
    }
}

__global__ void gat_elu(float* __restrict__ p, size_t n)
{
    size_t i = (size_t)blockIdx.x * blockDim.x + threadIdx.x;
    if (i < n) { float x = p[i]; p[i] = x > 0.0f ? x : expm1f(x); }
}

// L2 row-normalize h2 (30 cols) in place + copy to d_out; one wave per node
__global__ void gat_l2norm(float* __restrict__ h2, float* __restrict__ outD)
{
    int gid  = blockIdx.x * blockDim.x + threadIdx.x;
    int wid  = gid >> 5;
    int lane = gid & 31;
    if (wid >= N_NODES) return;
    float v = (lane < F2) ? h2[(size_t)wid * F2 + lane] : 0.0f;
    float sq = v * v;
    for (int o = 16; o > 0; o >>= 1) sq += __shfl_xor(sq, o, 32);
    float inv = 1.0f / fmaxf(sqrtf(sq), EPSV);
    if (lane < F2) {
        float r = v * inv;
        h2[(size_t)wid * F2 + lane]   = r;
        outD[(size_t)wid * F2 + lane] = r;
    }
}

// ---------------------------------------------------------------------------
extern "C" void kernel_launch(void* const* d_in, const int* in_sizes, int n_in,
                              void* d_out, int out_size, void* d_ws, size_t ws_size,
                              hipStream_t stream)
{
    const float* feat = (const float*)d_in[0];   // [20000,3000]
    const float* W1   = (const float*)d_in[1];   // [3000,512]
    const float* attS = (const float*)d_in[2];   // [512]
    const float* attD = (const float*)d_in[3];   // [512]
    const float* W2   = (const float*)d_in[4];   // [512,30]
    const int*   ei   = (const int*)  d_in[5];   // [2,160000]

    float* out_h2 = (float*)d_out;                       // [20000,30]
    float* out_h4 = out_h2 + (size_t)N_NODES * F2;       // [20000,3000]

    // workspace layout (floats)
    float* ws     = (float*)d_ws;
    float* x1     = ws;                                  // [20000,512]; reused as x3
    float* hbuf   = x1   + (size_t)N_NODES * F1;         // [20000,512]; h1 then h3
    float* h2n    = hbuf + (size_t)N_NODES * F1;         // [20000,30]
    float* aS     = h2n  + (size_t)N_NODES * F2;
    float* aD     = aS   + N_NODES;
    float* alpha  = aD   + N_NODES;                      // [180000]
    float* segmax = alpha + E_TOT;
    float* segsum = segmax + N_NODES;

    const int MT = (N_NODES + 63) / 64;                  // 313 M-tiles

    // conv1 linear: x1 = features @ W1   (M=20000, N=512, K=3000)
    {
        dim3 g((F1 + 127) / 128, MT);
        gat_gemm_wmma<false><<<g, 256, 0, stream>>>(feat, W1, x1, N_NODES, F1, IN_DIM);
    }
    // per-node attention logits
    gat_node_dots<<<(N_NODES * 32 + 255) / 256, 256, 0, stream>>>(x1, attS, attD, aS, aD);

    // segment softmax over dst (computed once, reused by conv1 AND conv3)
    gat_fill_f32<<<(N_NODES + 255) / 256, 256, 0, stream>>>(segmax, -INFINITY, N_NODES);
    hipMemsetAsync(segsum, 0, (size_t)N_NODES * sizeof(float), stream);
    const int EB = (E_TOT + 255) / 256;
    gat_edge_logit_max<<<EB, 256, 0, stream>>>(ei, aS, aD, alpha, segmax);
    gat_edge_exp_sum  <<<EB, 256, 0, stream>>>(ei, alpha, segmax, segsum);
    gat_edge_alpha    <<<EB, 256, 0, stream>>>(ei, alpha, segsum);

    // conv1 propagate + ELU -> h1
    hipMemsetAsync(hbuf, 0, (size_t)N_NODES * F1 * sizeof(float), stream);
    gat_propagate<<<(E_TOT * 32 + 255) / 256, 256, 0, stream>>>(ei, x1, alpha, hbuf);
    gat_elu<<<(int)(((size_t)N_NODES * F1 + 255) / 256), 256, 0, stream>>>(hbuf, (size_t)N_NODES * F1);

    // conv2: h2 = normalize(h1 @ W2)   (M=20000, N=30, K=512)
    {
        dim3 g(1, MT);
        gat_gemm_wmma<false><<<g, 256, 0, stream>>>(hbuf, W2, h2n, N_NODES, F2, F1);
    }
    gat_l2norm<<<(N_NODES * 32 + 255) / 256, 256, 0, stream>>>(h2n, out_h2);

    // conv3 linear: x3 = h2 @ W2^T  (NT: B = W2 [512,30]; M=20000, N=512, K=30)
    {
        dim3 g((F1 + 127) / 128, MT);
        gat_gemm_wmma<true><<<g, 256, 0, stream>>>(h2n, W2, x1, N_NODES, F1, F2);
    }

    // conv3 propagate (same alpha) + ELU -> h3
    hipMemsetAsync(hbuf, 0, (size_t)N_NODES * F1 * sizeof(float), stream);
    gat_propagate<<<(E_TOT * 32 + 255) / 256, 256, 0, stream>>>(ei, x1, alpha, hbuf);
    gat_elu<<<(int)(((size_t)N_NODES * F1 + 255) / 256), 256, 0, stream>>>(hbuf, (size_t)N_NODES * F1);

    // conv4: h4 = h3 @ W1^T  (NT: B = W1 [3000,512]; M=20000, N=3000, K=512)
    {
        dim3 g((IN_DIM + 127) / 128, MT);
        gat_gemm_wmma<true><<<g, 256, 0, stream>>>(hbuf, W1, out_h4, N_NODES, IN_DIM, F1);
    }
}